// MambaSSM_44942537786211
// MI455X (gfx1250) — compile-verified
//
#include <hip/hip_runtime.h>
#include <hip/hip_bf16.h>
#include <math.h>

// ---- problem dims (fixed by reference) ----
#define T_SEQ   1024
#define DM      513      // d_model
#define DI      1026     // d_inner
#define DS      16       // d_state
#define DC      4        // d_conv
#define DTR     33       // dt_rank
#define NL      4        // layers
#define DBLW    (DTR + 2*DS)   // 65
#define KB      64       // K-chunk staged in LDS

typedef __attribute__((ext_vector_type(2))) float v2f;
typedef __attribute__((ext_vector_type(8))) float v8f;

// -------------------------------------------------------------------
// x[t][f] = Z[f][t]
__global__ void mamba_transpose_kernel(const float* __restrict__ Z,
                                       float* __restrict__ X) {
  int idx = blockIdx.x * blockDim.x + threadIdx.x;
  if (idx >= T_SEQ * DM) return;
  int t = idx / DM, f = idx % DM;
  X[idx] = Z[(size_t)f * T_SEQ + t];
}

// -------------------------------------------------------------------
// O[M x N] = X[M x K] (row stride lda) @ W[N x K]^T  (+bias, +softplus)
// Block = 8 waves sharing one 16-row A panel (blockIdx.y selects tm).
// A panel is staged per 64-K chunk into LDS with async global->LDS copies
// (ASYNCcnt path), consumed as ds_load; each wave computes a 16x64 strip
// of N with 4 fp32-WMMA accumulators sharing the A operand.
__global__ __launch_bounds__(256)
void mamba_gemm_wmma_f32(const float* __restrict__ X, int lda,
                         const float* __restrict__ W,
                         const float* __restrict__ bias,
                         float* __restrict__ O,
                         int M, int N, int K, int act) {
  __shared__ float a_s[16 * KB];                 // 4 KB A-panel chunk

  const int tid  = threadIdx.x;
  const int lane = tid & 31;
  const int wv   = tid >> 5;
  const int tm   = blockIdx.y;                   // 16-row panel index
  const int sn   = blockIdx.x * 8 + wv;          // 64-wide N strip index

  const int r  = lane & 15;                      // row (A) / col (B)
  const int kb = (lane >> 4) << 1;               // k base for this half-wave

  const float* __restrict__ xpanel = X + (size_t)(tm * 16) * lda;
  const float* __restrict__ xrow   = xpanel + (size_t)r * lda;
  const float* wrow[4];
  int ncol[4];
#pragma unroll
  for (int j = 0; j < 4; ++j) {
    const int n = sn * 64 + j * 16 + r;
    ncol[j] = n;
    // Clamp to a valid row: out-of-range columns accumulate garbage in
    // lanes that are never stored; loads stay unconditional (EXEC all-1s).
    wrow[j] = W + (size_t)(n < N ? n : (N - 1)) * K;
  }

  // LDS byte offset of a_s: flat LDS aperture keeps the offset in addr[31:0].
  const unsigned lds_base = (unsigned)(uintptr_t)(void*)a_s;

  v8f c[4] = {v8f{}, v8f{}, v8f{}, v8f{}};
  const int K4 = K & ~3;

  for (int k0c = 0; k0c < K4; k0c += KB) {
    // ---- cooperative async copy: 16 rows x KB floats, 4 per thread ----
#pragma unroll
    for (int i = 0; i < 4; ++i) {
      const int e   = i * 256 + tid;             // 0..1023
      const int row = e >> 6;
      const int kk  = e & (KB - 1);
      const int kg  = k0c + kk;
      const float* gp = xpanel + (size_t)row * lda + (kg < K ? kg : (K - 1));
      const unsigned loff = lds_base + (unsigned)e * 4u;
      asm volatile("global_load_async_to_lds_b32 %0, %1, off"
                   :: "v"(loff), "v"(gp) : "memory");
    }
    asm volatile("s_wait_asynccnt 0x0" ::: "memory");
    __syncthreads();                             // panel chunk visible to all

    const int klim = (K4 - k0c < KB) ? (K4 - k0c) : KB;
    for (int kk = 0; kk < klim; kk += 4) {
      const int ka = kk + kb;                    // within chunk
      const int kg = k0c + ka;                   // global k
      __builtin_prefetch(&wrow[0][kg + KB], 0, 1);
      __builtin_prefetch(&wrow[2][kg + KB], 0, 1);
      v2f a;
      a.x = a_s[r * KB + ka];
      a.y = a_s[r * KB + ka + 1];
#pragma unroll
      for (int j = 0; j < 4; ++j) {
        v2f b;
        b.x = wrow[j][kg];
        b.y = wrow[j][kg + 1];
        c[j] = __builtin_amdgcn_wmma_f32_16x16x4_f32(false, a, false, b,
                                                     (short)0, c[j], false, false);
      }
    }
    __syncthreads();                             // done reading before refill
  }

  if (K4 < K) {                                  // K tail: value-masked, global A
    const int ka = K4 + kb, kc = ka + 1;
    const int kaC = (ka < K) ? ka : (K - 1);
    const int kcC = (kc < K) ? kc : (K - 1);
    v2f a;
    const float ax = xrow[kaC], ay = xrow[kcC];
    a.x = (ka < K) ? ax : 0.f;
    a.y = (kc < K) ? ay : 0.f;
#pragma unroll
    for (int j = 0; j < 4; ++j) {
      v2f b;
      const float bx = wrow[j][kaC], by = wrow[j][kcC];
      b.x = (ka < K) ? bx : 0.f;
      b.y = (kc < K) ? by : 0.f;
      c[j] = __builtin_amdgcn_wmma_f32_16x16x4_f32(false, a, false, b,
                                                   (short)0, c[j], false, false);
    }
  }

  const int half = lane >> 4;
#pragma unroll
  for (int j = 0; j < 4; ++j) {
    const int n = ncol[j];
    if (n < N) {
      const float bv = bias ? bias[n] : 0.f;
#pragma unroll
      for (int v = 0; v < 8; ++v) {
        const int mr = tm * 16 + v + 8 * half;
        float val = c[j][v] + bv;
        if (act == 1) {                          // softplus
          val = (val > 20.f) ? val : log1pf(__expf(val));
        }
        O[(size_t)mr * N + n] = val;
      }
    }
  }
}

// -------------------------------------------------------------------
// xa = silu( causal depthwise conv1d( xz[:, :DI] ) )
__global__ void mamba_conv_silu_kernel(const float* __restrict__ xz,
                                       const float* __restrict__ cw,
                                       const float* __restrict__ cb,
                                       float* __restrict__ xa) {
  int idx = blockIdx.x * blockDim.x + threadIdx.x;
  if (idx >= T_SEQ * DI) return;
  int t = idx / DI, ch = idx % DI;
  float acc = cb[ch];
#pragma unroll
  for (int k = 0; k < DC; ++k) {
    int tt = t + k - (DC - 1);
    if (tt >= 0) acc += xz[(size_t)tt * (2 * DI) + ch] * cw[ch * DC + k];
  }
  xa[idx] = acc / (1.f + __expf(-acc));          // silu
}

// -------------------------------------------------------------------
// Selective scan: 16 lanes per channel (one lane per state n).
// h = exp(dt*A)*h + dt*xa*B ; y_t = (sum_n h*C) + D*xa, gated by silu(z)
__global__ __launch_bounds__(256)
void mamba_scan_kernel(const float* __restrict__ dt,
                       const float* __restrict__ xa,
                       const float* __restrict__ dbl,
                       const float* __restrict__ A_log,
                       const float* __restrict__ Dp,
                       const float* __restrict__ z,   // = xz + DI, stride 2*DI
                       float* __restrict__ y) {
  int gid = blockIdx.x * blockDim.x + threadIdx.x;
  int ch = gid >> 4;
  int n  = gid & 15;
  if (ch >= DI) return;
  const float A  = -__expf(A_log[(size_t)ch * DS + n]);
  const float Dv = Dp[ch];
  float h = 0.f;
  for (int t = 0; t < T_SEQ; ++t) {
    const float dtv = dt[(size_t)t * DI + ch];
    const float xv  = xa[(size_t)t * DI + ch];
    const float Bv  = dbl[(size_t)t * DBLW + DTR + n];
    const float Cv  = dbl[(size_t)t * DBLW + DTR + DS + n];
    h = __expf(dtv * A) * h + dtv * xv * Bv;
    float p = h * Cv;
    // reduce the 16 states living in one 16-lane group
    for (int off = 8; off > 0; off >>= 1) p += __shfl_xor(p, off, 16);
    if (n == 0) {
      const float zv = z[(size_t)t * (2 * DI) + ch];
      const float sz = zv / (1.f + __expf(-zv));
      y[(size_t)t * DI + ch] = (p + Dv * xv) * sz;
    }
  }
}

// -------------------------------------------------------------------
__global__ void mamba_add_kernel(float* __restrict__ x,
                                 const float* __restrict__ o, int n) {
  int i = blockIdx.x * blockDim.x + threadIdx.x;
  if (i < n) x[i] += o[i];
}

// -------------------------------------------------------------------
__global__ __launch_bounds__(256)
void mamba_layernorm_kernel(float* __restrict__ x,
                            const float* __restrict__ g,
                            const float* __restrict__ b) {
  __shared__ float s1[256], s2[256];
  float* row = x + (size_t)blockIdx.x * DM;
  float a1 = 0.f, a2 = 0.f;
  for (int f = threadIdx.x; f < DM; f += 256) {
    float v = row[f]; a1 += v; a2 += v * v;
  }
  s1[threadIdx.x] = a1; s2[threadIdx.x] = a2;
  __syncthreads();
  for (int o = 128; o > 0; o >>= 1) {
    if (threadIdx.x < o) {
      s1[threadIdx.x] += s1[threadIdx.x + o];
      s2[threadIdx.x] += s2[threadIdx.x + o];
    }
    __syncthreads();
  }
  const float mu  = s1[0] / (float)DM;
  const float var = s2[0] / (float)DM - mu * mu;
  const float inv = rsqrtf(var + 1e-5f);
  for (int f = threadIdx.x; f < DM; f += 256)
    row[f] = (row[f] - mu) * inv * g[f] + b[f];
}

// -------------------------------------------------------------------
__global__ void mamba_grab_kernel(const float* __restrict__ x,
                                  float* __restrict__ out, int i) {
  if (threadIdx.x == 0 && blockIdx.x == 0)
    out[i] = x[(size_t)(T_SEQ - 1) * DM + (DM - 1)];
}

// ===================================================================
extern "C" void kernel_launch(void* const* d_in, const int* in_sizes, int n_in,
                              void* d_out, int out_size, void* d_ws, size_t ws_size,
                              hipStream_t stream) {
  const float* Z      = (const float*)d_in[0];
  const float* inpw   = (const float*)d_in[1];   // (L, 2*DI, DM)
  const float* convw  = (const float*)d_in[2];   // (L, DI, DC)
  const float* convb  = (const float*)d_in[3];   // (L, DI)
  const float* xpw    = (const float*)d_in[4];   // (L, 65, DI)
  const float* dtpw   = (const float*)d_in[5];   // (L, DI, DTR)
  const float* dtpb   = (const float*)d_in[6];   // (L, DI)
  const float* A_log  = (const float*)d_in[7];   // (L, DI, DS)
  const float* Dprm   = (const float*)d_in[8];   // (L, DI)
  const float* outw   = (const float*)d_in[9];   // (L, DM, DI)
  const float* ng     = (const float*)d_in[10];  // (L, DM)
  const float* nb     = (const float*)d_in[11];  // (L, DM)
  float* out = (float*)d_out;

  // ---- workspace carve-up (floats) ----
  float* ws = (float*)d_ws;
  float* x   = ws;                 ws += (size_t)T_SEQ * DM;        // 525,312
  float* xz  = ws;                 ws += (size_t)T_SEQ * 2 * DI;    // 2,101,248
  float* xa  = ws;                 ws += (size_t)T_SEQ * DI;        // 1,050,624
  float* dbl = ws;                 ws += (size_t)T_SEQ * DBLW;      //    66,560
  float* dt  = ws;                 ws += (size_t)T_SEQ * DI;        // 1,050,624
  float* yb  = ws;                 ws += (size_t)T_SEQ * DI;        // 1,050,624
  float* ob  = ws;                 ws += (size_t)T_SEQ * DM;        //   525,312
  (void)ws_size;

  const int nXM = T_SEQ * DM;
  const int nXI = T_SEQ * DI;

  mamba_transpose_kernel<<<(nXM + 255) / 256, 256, 0, stream>>>(Z, x);

  auto launch_gemm = [&](const float* Xp, int lda, const float* Wp,
                         const float* bp, float* Op,
                         int M, int N, int K, int act) {
    const int strips = (N + 63) >> 6;
    dim3 grid((strips + 7) / 8, M >> 4);
    mamba_gemm_wmma_f32<<<grid, 256, 0, stream>>>(Xp, lda, Wp, bp, Op,
                                                  M, N, K, act);
  };

  for (int l = 0; l < NL; ++l) {
    const float* W_in  = inpw  + (size_t)l * (2 * DI) * DM;
    const float* W_xp  = xpw   + (size_t)l * DBLW * DI;
    const float* W_dt  = dtpw  + (size_t)l * DI * DTR;
    const float* B_dt  = dtpb  + (size_t)l * DI;
    const float* W_out = outw  + (size_t)l * DM * DI;
    const float* CW    = convw + (size_t)l * DI * DC;
    const float* CB    = convb + (size_t)l * DI;
    const float* AL    = A_log + (size_t)l * DI * DS;
    const float* DP    = Dprm  + (size_t)l * DI;

    for (int rep = 0; rep < 2; ++rep) {
      // xz = x @ in_proj_w^T            (1024 x 2052)
      launch_gemm(x, DM, W_in, nullptr, xz, T_SEQ, 2 * DI, DM, 0);
      // xa = silu(conv1d(xi))
      mamba_conv_silu_kernel<<<(nXI + 255) / 256, 256, 0, stream>>>(xz, CW, CB, xa);
      // dbl = xa @ x_proj_w^T           (1024 x 65)
      launch_gemm(xa, DI, W_xp, nullptr, dbl, T_SEQ, DBLW, DI, 0);
      // dt = softplus(dbl[:, :33] @ dt_proj_w^T + b)   (1024 x 1026)
      launch_gemm(dbl, DBLW, W_dt, B_dt, dt, T_SEQ, DI, DTR, 1);
      // selective scan + D path + silu(z) gate -> y
      mamba_scan_kernel<<<(DI * DS + 255) / 256, 256, 0, stream>>>(
          dt, xa, dbl, AL, DP, xz + DI, yb);
      // out = y @ out_proj_w^T          (1024 x 513)
      launch_gemm(yb, DI, W_out, nullptr, ob, T_SEQ, DM, DI, 0);
      // x = out + residual(x)
      mamba_add_kernel<<<(nXM + 255) / 256, 256, 0, stream>>>(x, ob, nXM);
    }
    mamba_layernorm_kernel<<<T_SEQ, 256, 0, stream>>>(x, ng + (size_t)l * DM,
                                                      nb + (size_t)l * DM);
    mamba_grab_kernel<<<1, 32, 0, stream>>>(x, out, l);
  }
  (void)in_sizes; (void)n_in; (void)out_size;
}